// SMOTESAGE_82497731822016
// MI455X (gfx1250) — compile-verified
//
#include <hip/hip_runtime.h>
#include <hip/hip_bf16.h>

typedef float v2f __attribute__((ext_vector_type(2)));
typedef float v8f __attribute__((ext_vector_type(8)));

static __device__ __forceinline__ v8f wmma_f32_k4(v2f a, v2f b, v8f c) {
  // D = A(16x4 f32) * B(4x16 f32) + C(16x16 f32)
  return __builtin_amdgcn_wmma_f32_16x16x4_f32(false, a, false, b, (short)0, c,
                                               false, false);
}

__global__ void zero_kernel(float* __restrict__ p, long long count) {
  long long i = (long long)blockIdx.x * blockDim.x + threadIdx.x;
  long long stride = (long long)gridDim.x * blockDim.x;
  for (; i < count; i += stride) p[i] = 0.0f;
}

__global__ void degree_kernel(const int* __restrict__ dst, float* __restrict__ deg, int E) {
  int e = blockIdx.x * blockDim.x + threadIdx.x;
  if (e < E) unsafeAtomicAdd(&deg[dst[e]], 1.0f);
}

__global__ void invdeg_kernel(float* __restrict__ deg, int n) {
  int i = blockIdx.x * blockDim.x + threadIdx.x;
  if (i < n) deg[i] = 1.0f / fmaxf(deg[i], 1.0f);
}

// Layer-1 scatter: 16 threads per edge, one feature element each.
__global__ void scatter16_kernel(const int* __restrict__ src, const int* __restrict__ dst,
                                 const float* __restrict__ feat, float* __restrict__ agg,
                                 int E) {
  int gid = blockIdx.x * blockDim.x + threadIdx.x;
  int e = gid >> 4;
  int f = gid & 15;
  if (e < E) {
    int s = src[e], d = dst[e];
    unsafeAtomicAdd(&agg[(size_t)d * 16 + f], feat[(size_t)s * 16 + f]);
  }
}

// Layer-2 scatter: one wave per edge, float4 per lane (128 floats per row).
__global__ void scatter128_kernel(const int* __restrict__ src, const int* __restrict__ dst,
                                  const float* __restrict__ h, float* __restrict__ agg2,
                                  int E) {
  int gid = blockIdx.x * blockDim.x + threadIdx.x;
  int e = gid >> 5;
  int lane = threadIdx.x & 31;
  if (e < E) {
    int s = src[e], d = dst[e];
    float4 v = ((const float4*)(h + (size_t)s * 128))[lane];
    float* o = agg2 + (size_t)d * 128 + lane * 4;
    unsafeAtomicAdd(o + 0, v.x);
    unsafeAtomicAdd(o + 1, v.y);
    unsafeAtomicAdd(o + 2, v.z);
    unsafeAtomicAdd(o + 3, v.w);
  }
}

// Layer 1: h = relu( (agg1*invdeg) @ W1l + b1 + feat @ W1r ), one wave per 16-row tile.
__global__ void gemm1_wmma_kernel(const float* __restrict__ agg1, const float* __restrict__ feat,
                                  const float* __restrict__ invdeg,
                                  const float* __restrict__ W1l, const float* __restrict__ W1r,
                                  const float* __restrict__ b1, float* __restrict__ h, int n) {
  const int tile = blockIdx.x;
  const int lane = threadIdx.x & 31;
  const int M = lane & 15;            // row within tile (A), col within N-tile (B/C)
  const int koff = (lane >> 4) << 1;  // K offset: lanes 0-15 -> 0, 16-31 -> 2
  const int rbase = (lane >> 4) << 3; // C rows: lanes 0-15 -> M=r, 16-31 -> M=r+8
  const bool fullTile = (tile * 16 + 16) <= n; // wave-uniform
  int row = tile * 16 + M;
  int rowc = row < n ? row : n - 1;
  const float id = invdeg[rowc];
  const float* am = agg1 + (size_t)rowc * 16;
  const float* af = feat + (size_t)rowc * 16;

  v2f Am[4], Af[4];
#pragma unroll
  for (int c = 0; c < 4; ++c) {
    int k = 4 * c + koff;
    Am[c].x = am[k] * id;
    Am[c].y = am[k + 1] * id;
    Af[c].x = af[k];
    Af[c].y = af[k + 1];
  }

#pragma unroll
  for (int j = 0; j < 8; ++j) {        // 8 N-tiles of 16 (128 outputs)
    const int Ncol = j * 16 + M;
    const float bv = b1[Ncol];
    v8f acc;
#pragma unroll
    for (int r = 0; r < 8; ++r) acc[r] = bv;
#pragma unroll
    for (int c = 0; c < 4; ++c) {
      int k = 4 * c + koff;
      v2f Bl, Br;
      Bl.x = W1l[(size_t)k * 128 + Ncol];
      Bl.y = W1l[(size_t)(k + 1) * 128 + Ncol];
      Br.x = W1r[(size_t)k * 128 + Ncol];
      Br.y = W1r[(size_t)(k + 1) * 128 + Ncol];
      acc = wmma_f32_k4(Am[c], Bl, acc);
      acc = wmma_f32_k4(Af[c], Br, acc);
    }
    float* hp = h + (size_t)(tile * 16 + rbase) * 128 + Ncol;
    if (fullTile) {
#pragma unroll
      for (int r = 0; r < 8; ++r) hp[(size_t)r * 128] = fmaxf(acc[r], 0.0f);
    } else {
#pragma unroll
      for (int r = 0; r < 8; ++r) {
        if (tile * 16 + rbase + r < n) hp[(size_t)r * 128] = fmaxf(acc[r], 0.0f);
      }
    }
  }
}

// Layer 2: out = (agg2*invdeg) @ W2l + b2 + h @ W2r ; W2 is 128x3, padded to 16 cols.
__global__ void gemm2_wmma_kernel(const float* __restrict__ agg2, const float* __restrict__ h,
                                  const float* __restrict__ invdeg,
                                  const float* __restrict__ W2l, const float* __restrict__ W2r,
                                  const float* __restrict__ b2, float* __restrict__ out, int n) {
  const int tile = blockIdx.x;
  const int lane = threadIdx.x & 31;
  const int M = lane & 15;
  const int koff = (lane >> 4) << 1;
  const int rbase = (lane >> 4) << 3;
  const bool fullTile = (tile * 16 + 16) <= n; // wave-uniform
  int row = tile * 16 + M;
  int rowc = row < n ? row : n - 1;
  const float id = invdeg[rowc];
  const float* am = agg2 + (size_t)rowc * 128;
  const float* ah = h + (size_t)rowc * 128;
  const int Ncol = M;
  const bool active = (Ncol < 3);

  v8f acc;
  const float bv = active ? b2[Ncol] : 0.0f;
#pragma unroll
  for (int r = 0; r < 8; ++r) acc[r] = bv;

#pragma unroll 8
  for (int c = 0; c < 32; ++c) {       // K = 128 in chunks of 4
    int k = 4 * c + koff;
    v2f Am, Ah, Bl, Br;
    Am.x = am[k] * id;
    Am.y = am[k + 1] * id;
    Ah.x = ah[k];
    Ah.y = ah[k + 1];
    Bl.x = active ? W2l[(size_t)k * 3 + Ncol] : 0.0f;
    Bl.y = active ? W2l[(size_t)(k + 1) * 3 + Ncol] : 0.0f;
    Br.x = active ? W2r[(size_t)k * 3 + Ncol] : 0.0f;
    Br.y = active ? W2r[(size_t)(k + 1) * 3 + Ncol] : 0.0f;
    acc = wmma_f32_k4(Am, Bl, acc);
    acc = wmma_f32_k4(Ah, Br, acc);
  }

  if (active) {
    float* op = out + (size_t)(tile * 16 + rbase) * 3 + Ncol;
    if (fullTile) {
#pragma unroll
      for (int r = 0; r < 8; ++r) op[(size_t)r * 3] = acc[r];
    } else {
#pragma unroll
      for (int r = 0; r < 8; ++r) {
        if (tile * 16 + rbase + r < n) op[(size_t)r * 3] = acc[r];
      }
    }
  }
}

extern "C" void kernel_launch(void* const* d_in, const int* in_sizes, int n_in,
                              void* d_out, int out_size, void* d_ws, size_t ws_size,
                              hipStream_t stream) {
  const float* feature = (const float*)d_in[0];
  const int* ei = (const int*)d_in[1]; // [2, E] flat: row 0 = src, row 1 = dst
  // d_in[2] = edge_type (unused by the reference forward)
  const float* W1l = (const float*)d_in[3];
  const float* b1  = (const float*)d_in[4];
  const float* W1r = (const float*)d_in[5];
  const float* W2l = (const float*)d_in[6];
  const float* b2  = (const float*)d_in[7];
  const float* W2r = (const float*)d_in[8];
  float* out = (float*)d_out;

  const int n = in_sizes[0] / 16;
  const int E = in_sizes[1] / 2;
  const int* src = ei;
  const int* dst = ei + E;

  // Workspace: invdeg[n] | agg1[16n] | h[128n] | agg2[128n]
  float* invdeg = (float*)d_ws;
  float* agg1 = invdeg + n;
  float* h    = agg1 + (size_t)n * 16;
  float* agg2 = h + (size_t)n * 128;

  const int ntiles = (n + 15) / 16;

  // 1) zero deg + agg1 (17n floats)
  zero_kernel<<<2048, 256, 0, stream>>>(invdeg, (long long)n * 17);
  // 2) degree
  degree_kernel<<<(E + 255) / 256, 256, 0, stream>>>(dst, invdeg, E);
  // 3) invdeg = 1/max(deg,1)
  invdeg_kernel<<<(n + 255) / 256, 256, 0, stream>>>(invdeg, n);
  // 4) layer-1 scatter-add (16 threads per edge)
  {
    long long threads = (long long)E * 16;
    scatter16_kernel<<<(unsigned)((threads + 255) / 256), 256, 0, stream>>>(src, dst, feature, agg1, E);
  }
  // 5) layer-1 fused GEMM + bias + relu (WMMA f32)
  gemm1_wmma_kernel<<<ntiles, 32, 0, stream>>>(agg1, feature, invdeg, W1l, W1r, b1, h, n);
  // 6) zero agg2 (128n floats)
  zero_kernel<<<4096, 256, 0, stream>>>(agg2, (long long)n * 128);
  // 7) layer-2 scatter-add (one wave per edge)
  scatter128_kernel<<<(E + 7) / 8, 256, 0, stream>>>(src, dst, h, agg2, E);
  // 8) layer-2 fused GEMM + bias (WMMA f32)
  gemm2_wmma_kernel<<<ntiles, 32, 0, stream>>>(agg2, h, invdeg, W2l, W2r, b2, out, n);
}